// Model_24232205484605
// MI455X (gfx1250) — compile-verified
//
#include <hip/hip_runtime.h>

// CDNA5 / gfx1250 N-body all-pairs gravity using V_WMMA_F32_16X16X4_F32.
//
//   dist_sq[i][j] = |r_i|^2 + |r_j|^2 - 2 r_i . r_j + eps^2
// computed as a single 16x16x4 f32 WMMA per 16x16 tile:
//   A row i = ( x_i, y_i, z_i, 1 )
//   B col j = (-2x_j, -2y_j, -2z_j, |r_j|^2 + eps^2 )
//   C[v]    = |r_i(m)|^2   (folded into the WMMA accumulator, m = v + 8*half)
// so D = dist_sq directly.  Acceleration uses the expansion
//   a_i = sum_j w_ij r_j - (sum_j w_ij) r_i ,  w_ij = m_j * dist_sq^{-3/2}
// so no per-pair difference vectors are ever formed.

typedef __attribute__((ext_vector_type(2))) float v2f;
typedef __attribute__((ext_vector_type(8))) float v8f;

#define TILE    16
#define JSPLIT  16
#define SOFT2   0.0001f   // 0.01^2

// Raw ds_swizzle_b32 XOR butterfly within 32 lanes (and_mask=0x1f, or_mask=0).
template <int XMASK>
__device__ __forceinline__ float swz_xor_add(float v) {
    const int off = (XMASK << 10) | 0x1f;
    return v + __int_as_float(__builtin_amdgcn_ds_swizzle(__float_as_int(v), off));
}
__device__ __forceinline__ float half_reduce(float v) {
    v = swz_xor_add<1>(v);
    v = swz_xor_add<2>(v);
    v = swz_xor_add<4>(v);
    v = swz_xor_add<8>(v);
    return v;   // lanes 0-15 and 16-31 reduced independently (rows are disjoint per half)
}

__global__ __launch_bounds__(32)
void nbody_wmma_f32(const float* __restrict__ pos,
                    const float* __restrict__ mass,
                    float* __restrict__ out,
                    int n)
{
    const int tid   = threadIdx.x;   // 0..31 (wave32)
    const int half  = tid >> 4;      // which 16-lane half
    const int ln    = tid & 15;      // lane within half
    const int iBase = blockIdx.x * TILE;

    // ---- A fragment (16x4 f32): lane holds components (2*half, 2*half+1)
    //      of row i = iBase + ln, row vector (x, y, z, 1).
    const int ia = (iBase + ln) * 3;
    const float xi = pos[ia + 0];
    const float yi = pos[ia + 1];
    const float zi = pos[ia + 2];
    v2f a;
    a.x = half ? zi   : xi;
    a.y = half ? 1.0f : yi;

    // ---- C accumulator seed: |r_i|^2 for the 8 rows this half owns
    //      (C/D layout: VGPR v -> row m = v + 8*half, col = ln).
    v8f cinit;
#pragma unroll
    for (int v = 0; v < 8; ++v) {
        const int im = (iBase + 8 * half + v) * 3;
        const float px = pos[im + 0];
        const float py = pos[im + 1];
        const float pz = pos[im + 2];
        cinit[v] = fmaf(px, px, fmaf(py, py, pz * pz));
    }

    float accw[8], accx[8], accy[8], accz[8];
#pragma unroll
    for (int v = 0; v < 8; ++v) { accw[v] = 0.f; accx[v] = 0.f; accy[v] = 0.f; accz[v] = 0.f; }

    // ---- sweep this block's slice of j-tiles, one-deep load pipeline
    const int jTiles     = n / TILE;
    const int jtPerSplit = jTiles / JSPLIT;
    const int jt0        = blockIdx.y * jtPerSplit;
    const int jtEnd      = jt0 + jtPerSplit;

    int jp = jt0 * TILE + ln;
    float xjn = pos[jp * 3 + 0];
    float yjn = pos[jp * 3 + 1];
    float zjn = pos[jp * 3 + 2];
    float mjn = mass[jp];

    for (int jt = jt0; jt < jtEnd; ++jt) {
        // current tile's j-data (already in registers)
        const float xj = xjn, yj = yjn, zj = zjn, mj = mjn;

        // issue next tile's loads before the compute chain
        const int jn  = (jt + 1 < jtEnd) ? (jt + 1) : jt0;
        const int jnp = jn * TILE + ln;
        xjn = pos[jnp * 3 + 0];
        yjn = pos[jnp * 3 + 1];
        zjn = pos[jnp * 3 + 2];
        mjn = mass[jnp];

        const float rj2e = fmaf(xj, xj, fmaf(yj, yj, zj * zj)) + SOFT2;

        // B fragment (4x16 f32): VGPR0 = row 2*half, VGPR1 = row 2*half+1.
        v2f b;
        b.x = half ? (-2.0f * zj) : (-2.0f * xj);
        b.y = half ? rj2e         : (-2.0f * yj);

        // D = A*B + C  ==  softened dist^2 for 256 pairs
        const v8f d = __builtin_amdgcn_wmma_f32_16x16x4_f32(
                /*neg_a=*/false, a, /*neg_b=*/false, b,
                /*c_mod=*/(short)0, cinit, /*reuse_a=*/false, /*reuse_b=*/false);

        // Per lane: 8 rows (m = 8*half + v), one fixed column j.
#pragma unroll
        for (int v = 0; v < 8; ++v) {
            const float t = __builtin_amdgcn_rsqf(d[v]); // v_rsq_f32
            const float w = mj * (t * t * t);            // m_j * d2^-1.5
            accw[v] += w;
            accx[v]  = fmaf(w, xj, accx[v]);
            accy[v]  = fmaf(w, yj, accy[v]);
            accz[v]  = fmaf(w, zj, accz[v]);
        }
    }

    // ---- reduce across the 16 lanes of each half
#pragma unroll
    for (int v = 0; v < 8; ++v) {
        accw[v] = half_reduce(accw[v]);
        accx[v] = half_reduce(accx[v]);
        accy[v] = half_reduce(accy[v]);
        accz[v] = half_reduce(accz[v]);
    }

    // ---- lanes 0..7 of each half emit this tile's partial a_i for 16 rows
    if (ln < 8) {
        const int i  = iBase + 8 * half + ln;
        const int ib = i * 3;
        const float px = pos[ib + 0];
        const float py = pos[ib + 1];
        const float pz = pos[ib + 2];
        const float w  = accw[ln];
        unsafeAtomicAdd(&out[ib + 0], fmaf(-w, px, accx[ln]));
        unsafeAtomicAdd(&out[ib + 1], fmaf(-w, py, accy[ln]));
        unsafeAtomicAdd(&out[ib + 2], fmaf(-w, pz, accz[ln]));
    }
}

extern "C" void kernel_launch(void* const* d_in, const int* in_sizes, int n_in,
                              void* d_out, int out_size, void* d_ws, size_t ws_size,
                              hipStream_t stream)
{
    const float* pos  = (const float*)d_in[0];   // (N,3) f32
    const float* mass = (const float*)d_in[1];   // (N,)  f32
    float*       out  = (float*)d_out;           // (N,3) f32
    const int n = in_sizes[1];                   // N = 8192

    // d_out is poisoned by the harness; atomics accumulate, so zero it first
    // (hipMemsetAsync is graph-capture safe).
    hipMemsetAsync(d_out, 0, (size_t)out_size * sizeof(float), stream);

    dim3 grid(n / TILE, JSPLIT, 1);
    nbody_wmma_f32<<<grid, 32, 0, stream>>>(pos, mass, out, n);
}